// TtMambaSSM_18580028523056
// MI455X (gfx1250) — compile-verified
//
#include <hip/hip_runtime.h>
#include <hip/hip_bf16.h>

typedef __attribute__((ext_vector_type(2))) float v2f;
typedef __attribute__((ext_vector_type(8))) float v8f;

#define BATCH 256
#define DI    5120
#define RK    160
#define SN    16

// ---------------- kernel 0: negA = -exp(A_log), precomputed once ----------------
__global__ void k_negA(const float* __restrict__ A_log, float* __restrict__ negA) {
    int i = blockIdx.x * blockDim.x + threadIdx.x;
    if (i < DI * SN) negA[i] = -__expf(A_log[i]);
}

// ---------------- kernel 1: dtlow = x @ W_delta  (256x5120 @ 5120x160) ----------
// grid: (256/16)*(160/16)=160 blocks, 128 threads = 4 waves, K split across waves
__global__ void __launch_bounds__(128) k_dtlow(const float* __restrict__ X,
                                               const float* __restrict__ Wd,
                                               float* __restrict__ dtlow) {
    const int NW = 4;
    __shared__ float red[NW * 32 * 8];
    int tileM = blockIdx.x / (RK / 16);
    int tileN = blockIdx.x % (RK / 16);
    int lane  = threadIdx.x & 31;
    int w     = threadIdx.x >> 5;
    int m     = lane & 15;
    int kh    = lane >> 4;   // K-half select: V0 holds K = k+2*kh, V1 holds K = k+2*kh+1
    int n     = lane & 15;

    v8f c = {};
    const float* Arow = X + (tileM * 16 + m) * DI;
    const float* Bcol = Wd + tileN * 16 + n;
    int k0 = w * (DI / NW), k1 = k0 + DI / NW;
    for (int k = k0; k < k1; k += 4) {
        v2f a, b;
        a.x = Arow[k + 2 * kh + 0];
        a.y = Arow[k + 2 * kh + 1];
        b.x = Bcol[(k + 2 * kh + 0) * RK];
        b.y = Bcol[(k + 2 * kh + 1) * RK];
        c = __builtin_amdgcn_wmma_f32_16x16x4_f32(false, a, false, b, (short)0, c, false, false);
    }
    for (int r = 0; r < 8; ++r) red[(w * 32 + lane) * 8 + r] = c[r];
    __syncthreads();
    if (threadIdx.x < 32) {
        for (int r = 0; r < 8; ++r) {
            float s = 0.f;
            for (int ww = 0; ww < NW; ++ww) s += red[(ww * 32 + lane) * 8 + r];
            int mm = r + (lane >> 4) * 8;           // C/D layout: lanes 16-31 hold M+8
            dtlow[(tileM * 16 + mm) * RK + tileN * 16 + (lane & 15)] = s;
        }
    }
}

// ---------------- kernel 2: Bp = x @ W_B, C = x @ W_C (256x5120 @ 5120x16) ------
// grid: 32 blocks = {W_B, W_C} x 16 M-tiles; 8 waves split K=5120, LDS reduce
__global__ void __launch_bounds__(256) k_proj(const float* __restrict__ X,
                                              const float* __restrict__ WB,
                                              const float* __restrict__ WC,
                                              float* __restrict__ Bp,
                                              float* __restrict__ Cc) {
    const int NW = 8;
    __shared__ float red[NW * 32 * 8];
    int sel   = blockIdx.x & 1;
    int tileM = blockIdx.x >> 1;
    const float* W  = sel ? WC : WB;
    float*      Out = sel ? Cc : Bp;
    int lane = threadIdx.x & 31;
    int w    = threadIdx.x >> 5;
    int m = lane & 15, kh = lane >> 4, n = lane & 15;

    v8f c = {};
    const float* Arow = X + (tileM * 16 + m) * DI;
    int k0 = w * (DI / NW), k1 = k0 + DI / NW;
    for (int k = k0; k < k1; k += 4) {
        v2f a, b;
        a.x = Arow[k + 2 * kh + 0];
        a.y = Arow[k + 2 * kh + 1];
        b.x = W[(k + 2 * kh + 0) * SN + n];
        b.y = W[(k + 2 * kh + 1) * SN + n];
        c = __builtin_amdgcn_wmma_f32_16x16x4_f32(false, a, false, b, (short)0, c, false, false);
    }
    for (int r = 0; r < 8; ++r) red[(w * 32 + lane) * 8 + r] = c[r];
    __syncthreads();
    if (threadIdx.x < 32) {
        for (int r = 0; r < 8; ++r) {
            float s = 0.f;
            for (int ww = 0; ww < NW; ++ww) s += red[(ww * 32 + lane) * 8 + r];
            int mm = r + (lane >> 4) * 8;
            Out[(tileM * 16 + mm) * SN + (lane & 15)] = s;
        }
    }
}

// ---- per-4-state-channel update + readout: (exp(dt*A)*h + dt*x*B) . C ----------
__device__ __forceinline__ float ssm_grp(float dtv, float s, float4 Av,
                                         float4 hh, float4 bb, float4 cc) {
    float h0 = __expf(dtv * Av.x) * hh.x + s * bb.x;
    float h1 = __expf(dtv * Av.y) * hh.y + s * bb.y;
    float h2 = __expf(dtv * Av.z) * hh.z + s * bb.z;
    float h3 = __expf(dtv * Av.w) * hh.w + s * bb.w;
    return h0 * cc.x + h1 * cc.y + h2 * cc.z + h3 * cc.w;
}

// ---------------- kernel 3: fused dt-GEMM + softplus + state update + readout ----
// One wave owns a 16x16 dt tile (16 batch rows x 16 d cols):
//   c = dtlow-tile @ W_dt-tile (WMMA, K=160), z = c + b_dt, dt = softplus(z)
// then the same lanes stream the matching h[b,d,0:16] and produce y directly —
// the 5.2 MB dt tensor never touches HBM.
__global__ void __launch_bounds__(256) k_fused(const float* __restrict__ dtlow,
                                               const float* __restrict__ Wdt,
                                               const float* __restrict__ bdt,
                                               const float* __restrict__ X,
                                               const float* __restrict__ H,
                                               const float* __restrict__ negA,
                                               const float* __restrict__ Bp,
                                               const float* __restrict__ Cc,
                                               const float* __restrict__ Dw,
                                               float* __restrict__ Y) {
    int lane = threadIdx.x & 31;
    int w    = threadIdx.x >> 5;
    int t    = blockIdx.x * 8 + w;          // 5120 tiles = 16 Mtiles x 320 Ntiles
    int tileM = t / (DI / 16);
    int tileN = t % (DI / 16);
    int m = lane & 15, kh = lane >> 4, n = lane & 15;

    // --- WMMA: dt tile, K = RK ---
    v8f c = {};
    const float* Arow = dtlow + (tileM * 16 + m) * RK;
    const float* Bcol = Wdt + tileN * 16 + n;
    for (int k = 0; k < RK; k += 4) {
        v2f a, b;
        a.x = Arow[k + 2 * kh + 0];
        a.y = Arow[k + 2 * kh + 1];
        b.x = Bcol[(k + 2 * kh + 0) * DI];
        b.y = Bcol[(k + 2 * kh + 1) * DI];
        c = __builtin_amdgcn_wmma_f32_16x16x4_f32(false, a, false, b, (short)0, c, false, false);
    }

    // --- this lane's d column is shared by all 8 fragment rows ---
    int d = tileN * 16 + n;
    float bias = bdt[d];
    float Dv   = Dw[d];
    const float4* a4 = (const float4*)(negA) + d * (SN / 4);
    float4 A0 = a4[0], A1 = a4[1], A2 = a4[2], A3 = a4[3];

#pragma unroll
    for (int r = 0; r < 8; ++r) {
        int b = tileM * 16 + r + kh * 8;    // C/D layout: lanes 16-31 hold M+8
        float z   = c[r] + bias;
        float dtv = (z > 20.f) ? z : log1pf(__expf(z));   // softplus, thresh 20
        float xv  = X[b * DI + d];
        float s   = dtv * xv;
        const float4* h4 = (const float4*)(H)  + (size_t)(b * DI + d) * (SN / 4);
        const float4* b4 = (const float4*)(Bp) + b * (SN / 4);
        const float4* c4 = (const float4*)(Cc) + b * (SN / 4);
        float acc = ssm_grp(dtv, s, A0, h4[0], b4[0], c4[0])
                  + ssm_grp(dtv, s, A1, h4[1], b4[1], c4[1])
                  + ssm_grp(dtv, s, A2, h4[2], b4[2], c4[2])
                  + ssm_grp(dtv, s, A3, h4[3], b4[3], c4[3]);
        Y[b * DI + d] = acc + xv * Dv;
    }
}

extern "C" void kernel_launch(void* const* d_in, const int* in_sizes, int n_in,
                              void* d_out, int out_size, void* d_ws, size_t ws_size,
                              hipStream_t stream) {
    const float* x       = (const float*)d_in[0];
    const float* h       = (const float*)d_in[1];
    const float* W_delta = (const float*)d_in[2];
    const float* W_dt    = (const float*)d_in[3];
    const float* b_dt    = (const float*)d_in[4];
    const float* A_log   = (const float*)d_in[5];
    const float* W_B     = (const float*)d_in[6];
    const float* W_C     = (const float*)d_in[7];
    const float* Dw      = (const float*)d_in[8];
    float* y = (float*)d_out;

    char* ws = (char*)d_ws;
    float* negA  = (float*)ws; ws += (size_t)DI * SN * 4;      // 320 KB
    float* dtlow = (float*)ws; ws += (size_t)BATCH * RK * 4;   // 160 KB
    float* Bp    = (float*)ws; ws += (size_t)BATCH * SN * 4;   // 16 KB
    float* Cc    = (float*)ws;                                 // 16 KB

    k_negA <<<(DI * SN + 255) / 256, 256, 0, stream>>>(A_log, negA);
    k_dtlow<<<(BATCH / 16) * (RK / 16), 128, 0, stream>>>(x, W_delta, dtlow);
    k_proj <<<2 * (BATCH / 16), 256, 0, stream>>>(x, W_B, W_C, Bp, Cc);
    k_fused<<<(BATCH / 16) * (DI / 16) / 8, 256, 0, stream>>>(dtlow, W_dt, b_dt,
                                                              x, h, negA, Bp, Cc, Dw, y);
}